// KConvBNR_70970039599476
// MI455X (gfx1250) — compile-verified
//
#include <hip/hip_runtime.h>
#include <hip/hip_bf16.h>

typedef float v2f __attribute__((ext_vector_type(2)));
typedef float v8f __attribute__((ext_vector_type(8)));

// Problem constants
#define BB 4
#define CC 64
#define HH 96
#define WW 96
#define HW (HH * WW)            // 9216
#define NPIX (BB * CC * HW)     // 2359296
#define TH 24                   // output rows per block
#define TILES (HH / TH)         // 4
#define TROWS (TH + 2)          // 26 tile rows (halo)
#define TSTRIDE 104             // row stride; interior (gc=0..95) at +4 (16B aligned)
#define NTHREADS 384            // 96 cols x 4 row-groups of 6 rows
#define NWAVES (NTHREADS / 32)  // 12
#define NBLK (BB * CC * TILES)  // 1024

// Spline: knots t_i = (i-3)*0.4 - 1; tpos = 2.5*x + 5.5; interval j = floor(tpos) in [0,10].
// Nonzero cubic bases N_{j-3..j} with standard uniform polys of u = frac(tpos).
// Padded weight table pw[f][18]: pw[f][k] = (3<=k<=10) ? w_spline[f][k-3]*scaler[f] : 0.
// Gather index for term t is simply j+t; invalid j -> sentinel 14 (k=14..17 all zero).

struct Val { float s; float b0, b1, b2, b3; int j; };

__device__ __forceinline__ Val eval_val(float xf) {
    Val v;
    // silu on TRANS units: v_exp_f32 + v_rcp_f32
    float e  = __expf(-xf);
    v.s      = xf * __builtin_amdgcn_rcpf(1.0f + e);
    float tpos = fmaf(xf, 2.5f, 5.5f);
    float fj = floorf(tpos);
    int   j  = (int)fj;
    float u  = tpos - fj;
    float omu = 1.0f - u;
    float u2 = u * u, u3 = u2 * u;
    const float s6 = 0.16666667f;
    v.b0 = omu * omu * omu * s6;
    v.b1 = fmaf(3.0f, u3, fmaf(-6.0f, u2, 4.0f)) * s6;
    v.b2 = fmaf(-3.0f, u3, fmaf(3.0f, u2, fmaf(3.0f, u, 1.0f))) * s6;
    v.b3 = u3 * s6;
    v.j  = ((unsigned)j <= 10u) ? j : 14;   // sentinel lands in zero pad
    return v;
}

__global__ __launch_bounds__(NTHREADS) void kan_fwd_kernel(
    const float* __restrict__ x,
    const float* __restrict__ base_w,     // (9)
    const float* __restrict__ spline_w,   // (9,8) flat
    const float* __restrict__ spline_s,   // (9)
    float* __restrict__ y,                // d_out used as pre-BN scratch
    float* __restrict__ part_sum,         // (NBLK)
    float* __restrict__ part_sq)          // (NBLK)
{
    __shared__ __align__(16) float tile[TROWS * TSTRIDE];  // 26*104 = 2704 floats
    __shared__ float pw[9 * 18];                           // zero-padded spline weights
    __shared__ float wred[2 * NWAVES];                     // per-wave totals

    const int tid  = threadIdx.x;
    const int blk  = blockIdx.x;
    const int bc   = blk / TILES;
    const int trow = blk % TILES;
    const int r0   = trow * TH;
    const int c    = tid % 96;       // column
    const int rg   = tid / 96;       // row-group (0..3), 6 rows each
    const float* __restrict__ src = x + (size_t)bc * HW;

    // base weights via uniform scalar loads (stay in SGPRs)
    float wbr[9];
    #pragma unroll
    for (int f = 0; f < 9; ++f) wbr[f] = base_w[f];

    // build zero-padded spline weight table
    if (tid < 9 * 18) {
        int f = tid / 18, k = tid - f * 18;
        pw[tid] = (k >= 3 && k <= 10) ? spline_w[f * 8 + (k - 3)] * spline_s[f] : 0.0f;
    }

    // Targeted zeroing: compute only reads tile cols 3..100.
    // Async loads fill cols 4..99 of every in-image row, so zero:
    //  - halo columns 3 and 100 of all rows (conv zero-padding, left/right edge)
    if (tid < 2 * TROWS) {
        int rr = tid >> 1;
        tile[rr * TSTRIDE + ((tid & 1) ? 100 : 3)] = 0.0f;
    }
    //  - interior of top/bottom halo row when it falls outside the image
    if (trow == 0 && tid < 24)
        ((float4*)&tile[0 * TSTRIDE + 4])[tid] = make_float4(0.f, 0.f, 0.f, 0.f);
    if (trow == TILES - 1 && tid < 24)
        ((float4*)&tile[(TROWS - 1) * TSTRIDE + 4])[tid] = make_float4(0.f, 0.f, 0.f, 0.f);
    __syncthreads();

    // async global -> LDS halo tile, 128-bit ops (interior is 16B aligned)
    for (int i = tid; i < TROWS * 24; i += NTHREADS) {
        int rr  = i / 24;
        int seg = i - rr * 24;                 // 4-float segment 0..23
        int gr  = r0 - 1 + rr;
        if ((unsigned)gr < (unsigned)HH) {
            unsigned int lds_off = (unsigned int)(size_t)(&tile[rr * TSTRIDE + 4 + seg * 4]);
            unsigned long long ga = (unsigned long long)(size_t)(src + gr * WW + seg * 4);
            asm volatile("global_load_async_to_lds_b128 %0, %1, off"
                         :: "v"(lds_off), "v"(ga) : "memory");
        }
    }
    asm volatile("s_wait_asynccnt 0x0" ::: "memory");
    __syncthreads();

    // sliding 3x3 window of per-value state down the column
    const int rl0 = rg * 6;                   // first output row (local)
    Val win[3][3];

    #pragma unroll
    for (int r = 0; r < 2; ++r) {
        const float* tp = &tile[(rl0 + r) * TSTRIDE + c + 3];
        #pragma unroll
        for (int dc = 0; dc < 3; ++dc) win[r][dc] = eval_val(tp[dc]);
    }

    float lsum = 0.0f, lsq = 0.0f;
    #pragma unroll
    for (int m = 0; m < 6; ++m) {
        {   // bring in bottom row of the window
            const float* tp = &tile[(rl0 + m + 2) * TSTRIDE + c + 3];
            #pragma unroll
            for (int dc = 0; dc < 3; ++dc) win[(m + 2) % 3][dc] = eval_val(tp[dc]);
        }
        // 3 accumulators (one per tap row) -> FMA chain of 15 instead of 45
        float accr[3];
        #pragma unroll
        for (int dr = 0; dr < 3; ++dr) {
            const Val* vr = win[(m + dr) % 3];
            float a = 0.0f;
            #pragma unroll
            for (int dc = 0; dc < 3; ++dc) {
                const Val& v = vr[dc];
                const int f = dr * 3 + dc;
                a = fmaf(wbr[f], v.s, a);
                const int ai = f * 18 + v.j;  // ds_loads get immediate offsets for +0..+3
                a = fmaf(pw[ai],     v.b0, a);
                a = fmaf(pw[ai + 1], v.b1, a);
                a = fmaf(pw[ai + 2], v.b2, a);
                a = fmaf(pw[ai + 3], v.b3, a);
            }
            accr[dr] = a;
        }
        float acc = (accr[0] + accr[1]) + accr[2];   // fixed order -> deterministic

        y[(size_t)bc * HW + (size_t)(r0 + rl0 + m) * WW + c] = acc;
        lsum += acc;
        lsq   = fmaf(acc, acc, lsq);
    }

    // --- WMMA-based block reduction for BN statistics ---------------------
    // D = A(16x4) x ones(4x16): each wave reduces its 32 lane values with one
    // matrix op. Since B is all ones and we fold one full column of D, the result is
    // layout-independent: column sum of D = sum of all A entries = sum of lane values.
    // All 384 threads are active here -> EXEC is all 1s (WMMA requirement).
    {
        v2f av = {lsum, 0.0f};
        v2f aq = {lsq,  0.0f};
        v2f ones2 = {1.0f, 1.0f};
        v8f dsum = {};
        v8f dsq  = {};
        dsum = __builtin_amdgcn_wmma_f32_16x16x4_f32(false, av, false, ones2,
                                                     (short)0, dsum, false, false);
        dsq  = __builtin_amdgcn_wmma_f32_16x16x4_f32(false, aq, false, ones2,
                                                     (short)0, dsq,  false, false);
        float hs = ((dsum[0] + dsum[1]) + (dsum[2] + dsum[3]))
                 + ((dsum[4] + dsum[5]) + (dsum[6] + dsum[7]));
        float hq = ((dsq[0] + dsq[1]) + (dsq[2] + dsq[3]))
                 + ((dsq[4] + dsq[5]) + (dsq[6] + dsq[7]));
        hs += __shfl_xor(hs, 16);   // combine row halves (lanes 0-15 / 16-31)
        hq += __shfl_xor(hq, 16);

        const int wave = tid >> 5;
        const int lane = tid & 31;
        if (lane == 0) { wred[wave] = hs; wred[NWAVES + wave] = hq; }
        __syncthreads();
        if (tid == 0) {
            float s = 0.0f, q = 0.0f;
            #pragma unroll
            for (int w = 0; w < NWAVES; ++w) { s += wred[w]; q += wred[NWAVES + w]; }
            part_sum[blk] = s;
            part_sq[blk]  = q;
        }
    }
}

// One thread per channel: fold 16 partials in fixed order -> scale/shift
__global__ void bn_stats_kernel(const float* __restrict__ part_sum,
                                const float* __restrict__ part_sq,
                                const float* __restrict__ gamma,
                                const float* __restrict__ beta,
                                float* __restrict__ scsh)   // scale[64], shift[64]
{
    int c = threadIdx.x;
    if (c < CC) {
        float s = 0.0f, q = 0.0f;
        for (int b = 0; b < BB; ++b) {
            int base = (b * CC + c) * TILES;
            #pragma unroll
            for (int t = 0; t < TILES; ++t) {
                s += part_sum[base + t];
                q += part_sq[base + t];
            }
        }
        const float inv_n = 1.0f / (float)(BB * HW);
        float mean   = s * inv_n;
        float var    = fmaf(q, inv_n, -mean * mean);
        float invstd = rsqrtf(var + 1e-5f);
        float sc     = gamma[c] * invstd;
        scsh[c]      = sc;
        scsh[CC + c] = fmaf(-mean, sc, beta[c]);
    }
}

// In-place BN + ReLU on d_out, float4-vectorized
__global__ __launch_bounds__(256) void bn_apply_kernel(float4* __restrict__ y,
                                                       const float* __restrict__ scsh)
{
    int i = blockIdx.x * 256 + threadIdx.x;          // float4 index
    if (i < NPIX / 4) {
        int c = (i / (HW / 4)) & (CC - 1);
        float sc = scsh[c], sh = scsh[CC + c];
        float4 v = y[i];
        v.x = fmaxf(fmaf(v.x, sc, sh), 0.0f);
        v.y = fmaxf(fmaf(v.y, sc, sh), 0.0f);
        v.z = fmaxf(fmaf(v.z, sc, sh), 0.0f);
        v.w = fmaxf(fmaf(v.w, sc, sh), 0.0f);
        y[i] = v;
    }
}

extern "C" void kernel_launch(void* const* d_in, const int* in_sizes, int n_in,
                              void* d_out, int out_size, void* d_ws, size_t ws_size,
                              hipStream_t stream) {
    const float* x        = (const float*)d_in[0];
    const float* base_w   = (const float*)d_in[1];
    const float* spline_w = (const float*)d_in[2];
    const float* spline_s = (const float*)d_in[3];
    const float* gamma    = (const float*)d_in[4];
    const float* beta     = (const float*)d_in[5];

    float* y   = (float*)d_out;             // pre-BN scratch, finalized in-place
    float* wsf = (float*)d_ws;              // needs 2176 floats (~8.7 KB)
    float* part_sum = wsf;                  // [0, 1024)
    float* part_sq  = wsf + NBLK;           // [1024, 2048)
    float* scsh     = wsf + 2 * NBLK;       // [2048, 2176)

    kan_fwd_kernel<<<NBLK, NTHREADS, 0, stream>>>(x, base_w, spline_w, spline_s,
                                                  y, part_sum, part_sq);
    bn_stats_kernel<<<1, 64, 0, stream>>>(part_sum, part_sq, gamma, beta, scsh);
    bn_apply_kernel<<<NPIX / 4 / 256, 256, 0, stream>>>((float4*)y, scsh);
}